// Network_66795331388119
// MI455X (gfx1250) — compile-verified
//
#include <hip/hip_runtime.h>
#include <math.h>

// ---------------------------------------------------------------------------
// PointNet++ critic forward on MI455X (gfx1250, wave32).
// All GEMM-like layers use V_WMMA_F32_16X16X4_F32 (fp32 tensor path): the
// workload (~13 GFLOP, ~200 MB traffic, ~10us at 23.3 TB/s) is latency-bound,
// so fp32 fidelity costs nothing. Activations and packed weights use row
// strides padded to 16B so WMMA fragments load as unconditional b64s.
// ---------------------------------------------------------------------------

typedef float v2f __attribute__((ext_vector_type(2)));
typedef float v8f __attribute__((ext_vector_type(8)));

#define BB   8
#define NPTS 4096

static inline int round4(int c) { return (c + 3) & ~3; }

// -------------------------------- kernels ----------------------------------

__global__ void k_init_xyz(const float* __restrict__ pcs, const float* __restrict__ cp1,
                           float* __restrict__ xyz0)
{
  int i = blockIdx.x * blockDim.x + threadIdx.x;
  const int total = BB * NPTS * 3;
  if (i >= total) return;
  int c = i % 3; int r = i / 3; int n = r % NPTS; int b = r / NPTS;
  xyz0[i] = (n == 0) ? cp1[b * 3 + c] : pcs[i];
}

// Farthest-point sampling: one block per batch, dist lives in LDS.
__global__ void k_fps(const float* __restrict__ xyz, int n, int npoint,
                      int* __restrict__ out)
{
  __shared__ float dist[4096];
  __shared__ float rmax[256];
  __shared__ int   rarg[256];
  __shared__ float c[3];
  const int b = blockIdx.x;
  const float* p = xyz + (size_t)b * n * 3;
  const int t = threadIdx.x;
  for (int i = t; i < n; i += 256) dist[i] = 1e10f;
  __syncthreads();
  int far = 0;
  for (int j = 0; j < npoint; ++j) {
    if (t == 0) {
      out[b * npoint + j] = far;
      c[0] = p[far * 3 + 0]; c[1] = p[far * 3 + 1]; c[2] = p[far * 3 + 2];
    }
    __syncthreads();
    float bm = -1.f; int bi = n - 1;
    for (int i = t; i < n; i += 256) {
      float dx = p[i * 3 + 0] - c[0], dy = p[i * 3 + 1] - c[1], dz = p[i * 3 + 2] - c[2];
      float d  = dx * dx + dy * dy + dz * dz;
      float nd = fminf(dist[i], d);
      dist[i] = nd;
      if (nd > bm) { bm = nd; bi = i; }      // keeps first (smallest) index on ties
    }
    rmax[t] = bm; rarg[t] = bi;
    __syncthreads();
    for (int s = 128; s > 0; s >>= 1) {
      if (t < s) {
        if (rmax[t + s] > rmax[t] ||
            (rmax[t + s] == rmax[t] && rarg[t + s] < rarg[t])) {
          rmax[t] = rmax[t + s]; rarg[t] = rarg[t + s];
        }
      }
      __syncthreads();
    }
    far = rarg[0];
    __syncthreads();
  }
}

__global__ void k_gather_xyz(const float* __restrict__ xyz, const int* __restrict__ idx,
                             int n, int S, float* __restrict__ out)
{
  int i = blockIdx.x * blockDim.x + threadIdx.x;
  if (i >= BB * S) return;
  int b = i / S;
  int id = idx[i];
  const float* s = xyz + ((size_t)b * n + id) * 3;
  out[(size_t)i * 3 + 0] = s[0];
  out[(size_t)i * 3 + 1] = s[1];
  out[(size_t)i * 3 + 2] = s[2];
}

// Ball query: ascending index scan == reference's sort-then-truncate semantics.
__global__ void k_ball(const float* __restrict__ xyz, const float* __restrict__ nxz,
                       int n, int S, float r2, int* __restrict__ idx)
{
  int i = blockIdx.x * blockDim.x + threadIdx.x;
  if (i >= BB * S) return;
  int b = i / S;
  const float* q = nxz + (size_t)i * 3;
  const float* p = xyz + (size_t)b * n * 3;
  int* o = idx + (size_t)i * 32;
  int cnt = 0;
  for (int j = 0; j < n && cnt < 32; ++j) {
    float dx = p[j * 3 + 0] - q[0], dy = p[j * 3 + 1] - q[1], dz = p[j * 3 + 2] - q[2];
    if (dx * dx + dy * dy + dz * dz <= r2) o[cnt++] = j;
  }
  int f = cnt ? o[0] : 0;
  for (; cnt < 32; ++cnt) o[cnt] = f;
}

// Build grouped input rows (padded stride Cp): [rel_xyz(3) | feats(Cf) | 0 pad]
__global__ void k_group(const float* __restrict__ xyz, const float* __restrict__ feats,
                        const float* __restrict__ nxz, const int* __restrict__ idx,
                        int n, int S, int Cf, int Cp, float* __restrict__ X)
{
  const int Cin = 3 + Cf;
  long i = (long)blockIdx.x * blockDim.x + threadIdx.x;
  long total = (long)BB * S * 32 * Cp;
  if (i >= total) return;
  int c = (int)(i % Cp); long r = i / Cp;
  int k = (int)(r % 32); long bs = r / 32;
  int b = (int)(bs / S);
  float v = 0.f;
  if (c < Cin) {
    int id = idx[bs * 32 + k];
    if (c < 3) v = xyz[((size_t)b * n + id) * 3 + c] - nxz[bs * 3 + c];
    else       v = feats[((size_t)b * n + id) * Cf + (c - 3)];
  }
  X[i] = v;
}

// Pack weights O x C (stride C) -> O x Cp (stride Cp, zero pad).
__global__ void k_packW(const float* __restrict__ W, int O, int C, int Cp,
                        float* __restrict__ Wp)
{
  long i = (long)blockIdx.x * blockDim.x + threadIdx.x;
  long total = (long)O * Cp;
  if (i >= total) return;
  int kp = (int)(i % Cp); int o = (int)(i / Cp);
  Wp[i] = (kp < C) ? W[(size_t)o * C + kp] : 0.f;
}

// Y[r,o] = sum_c X[r,c]*Wp[o,c] + bias[o] via V_WMMA_F32_16X16X4_F32.
// 4 waves/block, one 16x16 D tile per wave. R,O multiples of 16; Cp % 4 == 0,
// rows 16B aligned -> unconditional aligned v2f (b64) fragment loads.
__global__ void __launch_bounds__(128)
k_gemm(const float* __restrict__ X, const float* __restrict__ Wp,
       const float* __restrict__ bias, float* __restrict__ Y,
       int R, int Cp, int O)
{
  const int lane = threadIdx.x & 31;
  const int wave = threadIdx.x >> 5;
  const int m0 = (blockIdx.x * 4 + wave) * 16;
  const int n0 = blockIdx.y * 16;
  if (m0 >= R) return;
  const int hi = lane >> 4;          // 0: K pair {0,1}; 1: K pair {2,3}
  const int l  = lane & 15;
  const float* xr = X  + (size_t)(m0 + l) * Cp + 2 * hi;  // A row  (M = l)
  const float* wr = Wp + (size_t)(n0 + l) * Cp + 2 * hi;  // B col  (N = l), B = W^T
  __builtin_prefetch(xr, 0, 3);                            // global_prefetch_b8
  __builtin_prefetch(wr, 0, 3);
  v8f acc = {0.f, 0.f, 0.f, 0.f, 0.f, 0.f, 0.f, 0.f};
#pragma unroll 2
  for (int k0 = 0; k0 < Cp; k0 += 4) {
    v2f a  = *(const v2f*)(xr + k0);
    v2f bf = *(const v2f*)(wr + k0);
    acc = __builtin_amdgcn_wmma_f32_16x16x4_f32(
        /*neg_a=*/false, a, /*neg_b=*/false, bf,
        /*c_mod=*/(short)0, acc, /*reuse_a=*/false, /*reuse_b=*/false);
  }
  const int oc = n0 + l;
  const float bv = bias ? bias[oc] : 0.f;
#pragma unroll
  for (int v = 0; v < 8; ++v) {
    int r = m0 + v + 8 * hi;                    // C/D layout: M = vgpr + 8*hi
    Y[(size_t)r * O + oc] = acc[v] + bv;
  }
}

// Deterministic per-channel sum / sumsq (one block per channel).
__global__ void k_stats(const float* __restrict__ Y, int R, int O,
                        float* __restrict__ st)
{
  __shared__ float s1[256], s2[256];
  int o = blockIdx.x; int t = threadIdx.x;
  float a = 0.f, b = 0.f;
  for (int r = t; r < R; r += 256) {
    float v = Y[(size_t)r * O + o];
    a += v; b += v * v;
  }
  s1[t] = a; s2[t] = b;
  __syncthreads();
  for (int s = 128; s > 0; s >>= 1) {
    if (t < s) { s1[t] += s1[t + s]; s2[t] += s2[t + s]; }
    __syncthreads();
  }
  if (t == 0) { st[o] = s1[0]; st[O + o] = s2[0]; }
}

__global__ void k_bn_relu(float* __restrict__ Y, const float* __restrict__ st,
                          const float* __restrict__ ga, const float* __restrict__ be,
                          int R, int O)
{
  long i = (long)blockIdx.x * blockDim.x + threadIdx.x;
  long total = (long)R * O;
  if (i >= total) return;
  int o = (int)(i % O);
  float inv = 1.f / (float)R;
  float m = st[o] * inv;
  float v = st[O + o] * inv - m * m;
  float y = ga[o] * (Y[i] - m) * rsqrtf(v + 1e-5f) + be[o];
  Y[i] = fmaxf(y, 0.f);
}

// out[b,s,o] = max_k Y[b,s,k,o]
__global__ void k_maxpool(const float* __restrict__ Y, int S, int O,
                          float* __restrict__ out)
{
  long i = (long)blockIdx.x * blockDim.x + threadIdx.x;
  long total = (long)BB * S * O;
  if (i >= total) return;
  int o = (int)(i % O); long bs = i / O;
  const float* src = Y + ((size_t)bs * 32) * O + o;
  float m = src[0];
  for (int k = 1; k < 32; ++k) m = fmaxf(m, src[(size_t)k * O]);
  out[i] = m;
}

// FP stage: 3-NN inverse-distance interpolation + concat [f1 | interp(f2) | 0 pad].
__global__ void k_fp_build(const float* __restrict__ x1, const float* __restrict__ x2,
                           const float* __restrict__ f1, const float* __restrict__ f2,
                           int N1, int S2, int C1, int C2, int Cp,
                           float* __restrict__ X)
{
  int i = blockIdx.x * blockDim.x + threadIdx.x;
  if (i >= BB * N1) return;
  int b = i / N1;
  const float* q = x1 + (size_t)i * 3;
  const float* p = x2 + (size_t)b * S2 * 3;
  float d0 = 3e38f, d1 = 3e38f, d2 = 3e38f; int i0 = 0, i1 = 0, i2 = 0;
  for (int j = 0; j < S2; ++j) {
    float dx = p[j * 3 + 0] - q[0], dy = p[j * 3 + 1] - q[1], dz = p[j * 3 + 2] - q[2];
    float d = dx * dx + dy * dy + dz * dz;
    if (d < d0)      { d2 = d1; i2 = i1; d1 = d0; i1 = i0; d0 = d; i0 = j; }
    else if (d < d1) { d2 = d1; i2 = i1; d1 = d;  i1 = j; }
    else if (d < d2) { d2 = d;  i2 = j; }
  }
  float w0 = 1.f / (d0 + 1e-8f), w1 = 1.f / (d1 + 1e-8f), w2 = 1.f / (d2 + 1e-8f);
  float ws = w0 + w1 + w2; w0 /= ws; w1 /= ws; w2 /= ws;
  float* o = X + (size_t)i * Cp;
  const float* a = f1 + (size_t)i * C1;
  for (int c = 0; c < C1; ++c) o[c] = a[c];
  const float* g0 = f2 + ((size_t)b * S2 + i0) * C2;
  const float* g1 = f2 + ((size_t)b * S2 + i1) * C2;
  const float* g2 = f2 + ((size_t)b * S2 + i2) * C2;
  for (int c = 0; c < C2; ++c) o[C1 + c] = w0 * g0[c] + w1 * g1[c] + w2 * g2[c];
  for (int c = C1 + C2; c < Cp; ++c) o[c] = 0.f;
}

// Critic head: net1(128) ++ cp_f(32) ++ dir_f(32) -> leaky_relu(W1x+b1) -> W2h+b2
__global__ void k_critic(const float* __restrict__ fcY,
                         const float* __restrict__ cp1, const float* __restrict__ dir1,
                         const float* __restrict__ cpW, const float* __restrict__ cpb,
                         const float* __restrict__ diW, const float* __restrict__ dib,
                         const float* __restrict__ W1, const float* __restrict__ b1,
                         const float* __restrict__ W2, const float* __restrict__ b2,
                         float* __restrict__ out)
{
  __shared__ float inp[192];
  __shared__ float h[128];
  int t = threadIdx.x;   // 128 threads
  for (int b = 0; b < BB; ++b) {
    inp[t] = fcY[((size_t)b * NPTS) * 128 + t];   // net1 = fc output at point 0
    if (t < 32) {
      const float* c = cp1 + b * 3;
      inp[128 + t] = cpW[t * 3] * c[0] + cpW[t * 3 + 1] * c[1] + cpW[t * 3 + 2] * c[2] + cpb[t];
      const float* d = dir1 + b * 3;
      inp[160 + t] = diW[t * 3] * d[0] + diW[t * 3 + 1] * d[1] + diW[t * 3 + 2] * d[2] + dib[t];
    }
    __syncthreads();
    float acc = b1[t];
    const float* wr = W1 + t * 192;
    for (int c = 0; c < 192; ++c) acc += wr[c] * inp[c];
    float hv = (acc > 0.f) ? acc : 0.01f * acc;   // leaky_relu(0.01)
    h[t] = hv * W2[t];
    __syncthreads();
    for (int s = 64; s > 0; s >>= 1) { if (t < s) h[t] += h[t + s]; __syncthreads(); }
    if (t == 0) out[b] = 0.1f / (1.f + __expf(-(h[0] + b2[0])));
    __syncthreads();
  }
}

// ------------------------------ host driver --------------------------------

extern "C" void kernel_launch(void* const* d_in, const int* in_sizes, int n_in,
                              void* d_out, int out_size, void* d_ws, size_t ws_size,
                              hipStream_t stream)
{
  (void)in_sizes; (void)n_in; (void)out_size; (void)ws_size;
  // Top level: setup_inputs() dict order.
  const float* pcs  = (const float*)d_in[0];
  const float* cp1  = (const float*)d_in[1];
  const float* dir1 = (const float*)d_in[2];
  int cur = 3;
  auto nxt = [&]() { return (const float*)d_in[cur++]; };
  // params pytree (dict keys sorted): cp, critic, dir, fc, fp, sa.
  const float* cpW  = nxt(); const float* cpb  = nxt();                 // cp: W, b
  const float* crW1 = nxt(); const float* crW2 = nxt();                 // critic: W1, W2,
  const float* crb1 = nxt(); const float* crb2 = nxt();                 //         b1, b2
  const float* diW  = nxt(); const float* dib  = nxt();                 // dir: W, b
  const float* fcW  = nxt(); const float* fcBe = nxt(); const float* fcGa = nxt(); // fc: W, beta, gamma
  struct Mlp { const float *g[3], *be[3], *W[3], *b[3]; };
  Mlp fp[4], sa[4];
  const int fpNL[4] = {3, 2, 2, 2};
  for (int i = 0; i < 4; ++i) {                                         // fp[i]: bns then convs
    for (int l = 0; l < fpNL[i]; ++l) { fp[i].g[l] = nxt(); fp[i].be[l] = nxt(); }
    for (int l = 0; l < fpNL[i]; ++l) { fp[i].W[l] = nxt(); fp[i].b[l]  = nxt(); }
  }
  for (int i = 0; i < 4; ++i) {                                         // sa[i]: bns then convs
    for (int l = 0; l < 3; ++l) { sa[i].g[l] = nxt(); sa[i].be[l] = nxt(); }
    for (int l = 0; l < 3; ++l) { sa[i].W[l] = nxt(); sa[i].b[l]  = nxt(); }
  }

  // Workspace carve-up (~170 MB).
  char* wp = (char*)d_ws;
  auto allocf = [&](size_t ne) { float* p = (float*)wp; wp += ((ne * 4 + 255) & ~(size_t)255); return p; };
  auto alloci = [&](size_t ne) { int*   p = (int*)wp;   wp += ((ne * 4 + 255) & ~(size_t)255); return p; };

  float* xyz0 = allocf((size_t)BB * NPTS * 3);
  const int SP[5] = {NPTS, 1024, 256, 64, 16};
  float* nxz[5];
  for (int i = 1; i <= 4; ++i) nxz[i] = allocf((size_t)BB * SP[i] * 3);
  int* fpsIdx  = alloci((size_t)BB * 1024);
  int* ballIdx = alloci((size_t)BB * 1024 * 32);
  float* Xbuf  = allocf(4700000);    // max padded rows*Cp (SA2: 65536*68 = 4.46M)
  float* Wpack = allocf(200704);     // max O*Cp = 256*768
  float* bufA  = allocf(17000000);   // max layer out: 262144x64
  float* bufB  = allocf(8500000);    // max alt out:  262144x32
  float* feats1 = allocf((size_t)BB * 1024 * 64);
  float* feats2 = allocf((size_t)BB * 256 * 128);
  float* feats3 = allocf((size_t)BB * 64 * 256);
  float* feats4 = allocf((size_t)BB * 16 * 512);
  float* nf3 = allocf((size_t)BB * 64 * 256);
  float* nf2 = allocf((size_t)BB * 256 * 256);
  float* nf1 = allocf((size_t)BB * 1024 * 128);
  float* nf0 = allocf((size_t)BB * NPTS * 128);
  float* fcY = allocf((size_t)BB * NPTS * 128);
  float* stats = allocf(1024);

  // One layer: pack W (stride C -> Cp), WMMA GEMM + bias, BN stats, BN+ReLU.
  // Xi must have row stride Cp (= round4(C)); C is the logical in-channels.
  auto gemm_bn_relu = [&](const float* Xi, const float* Wt, const float* bi,
                          const float* ga, const float* be, int R, int C, int O,
                          float* Yo) {
    int Cp = round4(C);
    long tw = (long)O * Cp;
    k_packW<<<(unsigned)((tw + 255) / 256), 256, 0, stream>>>(Wt, O, C, Cp, Wpack);
    dim3 g((unsigned)((R + 63) / 64), (unsigned)(O / 16)), blk(128);
    k_gemm<<<g, blk, 0, stream>>>(Xi, Wpack, bi, Yo, R, Cp, O);
    k_stats<<<O, 256, 0, stream>>>(Yo, R, O, stats);
    long tot = (long)R * O;
    k_bn_relu<<<(unsigned)((tot + 255) / 256), 256, 0, stream>>>(Yo, stats, ga, be, R, O);
  };

  auto run_mlp = [&](const float* Xi, int R, int C0, const Mlp& m, int nl,
                     const int* oc, float* finalDst) -> const float* {
    const float* curp = Xi; int C = C0;   // Xi stride = round4(C0)
    float* ping[2] = {bufA, bufB};
    const float* lastp = nullptr;
    for (int l = 0; l < nl; ++l) {
      float* target = (l == nl - 1 && finalDst) ? finalDst : ping[l & 1];
      gemm_bn_relu(curp, m.W[l], m.b[l], m.g[l], m.be[l], R, C, oc[l], target);
      curp = target; lastp = target; C = oc[l];   // oc multiples of 16 -> stride ok
    }
    return lastp;
  };

  auto sa_mod = [&](const float* xyzi, const float* fi, int n, int Cf, int S,
                    float radius, const Mlp& m, const int* oc,
                    float* nx, float* pooled) {
    k_fps<<<BB, 256, 0, stream>>>(xyzi, n, S, fpsIdx);
    int q = BB * S;
    k_gather_xyz<<<(q + 255) / 256, 256, 0, stream>>>(xyzi, fpsIdx, n, S, nx);
    k_ball<<<(q + 255) / 256, 256, 0, stream>>>(xyzi, nx, n, S, radius * radius, ballIdx);
    int Cp = round4(3 + Cf);
    long tg = (long)q * 32 * Cp;
    k_group<<<(unsigned)((tg + 255) / 256), 256, 0, stream>>>(xyzi, fi, nx, ballIdx, n, S, Cf, Cp, Xbuf);
    const float* y = run_mlp(Xbuf, q * 32, 3 + Cf, m, 3, oc, nullptr);
    long tp = (long)q * oc[2];
    k_maxpool<<<(unsigned)((tp + 255) / 256), 256, 0, stream>>>(y, S, oc[2], pooled);
  };

  auto fp_mod = [&](const float* x1, const float* x2, const float* f1, const float* f2,
                    int N1, int S2, int C1, int C2, const Mlp& m, int nl,
                    const int* oc, float* outp) {
    int q = BB * N1;
    int Cp = round4(C1 + C2);
    k_fp_build<<<(q + 127) / 128, 128, 0, stream>>>(x1, x2, f1, f2, N1, S2, C1, C2, Cp, Xbuf);
    run_mlp(Xbuf, q, C1 + C2, m, nl, oc, outp);
  };

  // ---- pipeline ----
  k_init_xyz<<<(BB * NPTS * 3 + 255) / 256, 256, 0, stream>>>(pcs, cp1, xyz0);

  const int sa_oc[4][3] = {{32,32,64},{64,64,128},{128,128,256},{256,256,512}};
  sa_mod(xyz0,   xyz0,   NPTS, 3,   1024, 0.1f, sa[0], sa_oc[0], nxz[1], feats1);
  sa_mod(nxz[1], feats1, 1024, 64,  256,  0.2f, sa[1], sa_oc[1], nxz[2], feats2);
  sa_mod(nxz[2], feats2, 256,  128, 64,   0.4f, sa[2], sa_oc[2], nxz[3], feats3);
  sa_mod(nxz[3], feats3, 64,   256, 16,   0.8f, sa[3], sa_oc[3], nxz[4], feats4);

  const int fp_oc[4][3] = {{128,128,128},{256,128,0},{256,256,0},{256,256,0}};
  fp_mod(nxz[3], nxz[4], feats3, feats4, 64,   16,   256, 512, fp[3], 2, fp_oc[3], nf3);
  fp_mod(nxz[2], nxz[3], feats2, nf3,    256,  64,   128, 256, fp[2], 2, fp_oc[2], nf2);
  fp_mod(nxz[1], nxz[2], feats1, nf2,    1024, 256,  64,  256, fp[1], 2, fp_oc[1], nf1);
  fp_mod(xyz0,   nxz[1], xyz0,   nf1,    NPTS, 1024, 3,   128, fp[0], 3, fp_oc[0], nf0);

  // fc: 128 -> 128, no bias, BN(gamma,beta) + relu
  gemm_bn_relu(nf0, fcW, nullptr, fcGa, fcBe, BB * NPTS, 128, 128, fcY);

  // critic head
  k_critic<<<1, 128, 0, stream>>>(fcY, cp1, dir1, cpW, cpb, diW, dib,
                                  crW1, crb1, crW2, crb2, (float*)d_out);
}